// LogitsMaskLayer_34720515620891
// MI455X (gfx1250) — compile-verified
//
#include <hip/hip_runtime.h>
#include <stdint.h>

#define SEP_TOKEN 2

// Native clang vector type: accepted by __builtin_nontemporal_{load,store}
// and lowers to single B128 global accesses.
typedef __attribute__((ext_vector_type(4))) float v4f;

// ---------------------------------------------------------------------------
// Kernel 1: pack word2syllables (int32, values 1..7) into one byte per word.
// 32 KB table -> stays resident in L2 / WGP$ for the streaming kernel.
// ---------------------------------------------------------------------------
__global__ void pack_w2s_kernel(const int* __restrict__ w2s,
                                uint8_t* __restrict__ packed, int V) {
    int i = blockIdx.x * blockDim.x + threadIdx.x;
    if (i < V) packed[i] = (uint8_t)w2s[i];
}

// ---------------------------------------------------------------------------
// Kernel 2: sequential syllable scan. One thread per batch column (B=128),
// S=32 steps. Tiny (4096 outputs), latency-bound but negligible vs. 1 GB
// of streaming below. remain values are always in [0,7] -> store as bytes.
// ---------------------------------------------------------------------------
__global__ void remain_scan_kernel(const int* __restrict__ dec,
                                   const int* __restrict__ w2s,
                                   uint8_t* __restrict__ remain,
                                   int S, int B) {
    int b = threadIdx.x + blockIdx.x * blockDim.x;
    if (b >= B) return;
    const int pattern[6] = {5, 7, 5, 7, 7, 0};

    int tok = dec[b];
    int seg = (tok == SEP_TOKEN) ? 1 : 0;
    int rem = (tok == SEP_TOKEN) ? pattern[1] : pattern[0];
    remain[b] = (uint8_t)rem;

    for (int s = 1; s < S; ++s) {
        tok = dec[s * B + b];
        rem -= w2s[tok];
        if (rem < 0) rem = 0;
        if (tok == SEP_TOKEN) {
            seg = (seg >= 5) ? 5 : seg + 1;
            rem = pattern[seg];
        }
        remain[s * B + b] = (uint8_t)rem;
    }
}

// ---------------------------------------------------------------------------
// Kernel 3: streaming mask. Pure bandwidth kernel:
//   16B nontemporal B128 load (logits) + 4B packed-byte load (L2-resident)
//   -> SWAR byte-compare vs wave-uniform remain -> 16B nontemporal B128 store.
// remain[blockIdx.y] is uniform per block (scalarizes to s_load).
// __builtin_prefetch emits global_prefetch_b8 to run the stream ~1 MB ahead.
// ---------------------------------------------------------------------------
__global__ void __launch_bounds__(256)
mask_kernel(const v4f* __restrict__ logits4,
            const uint32_t* __restrict__ pw2s4,
            const uint8_t* __restrict__ remain,
            v4f* __restrict__ out4,
            int V4, int rows) {
    const int row = blockIdx.y;
    const int v4  = blockIdx.x * blockDim.x + threadIdx.x;
    if (v4 >= V4) return;

    const size_t idx = (size_t)row * (size_t)V4 + (size_t)v4;

    // Prefetch the same column 8 rows (8*32000*4B = 1 MB) ahead, clamped
    // inside the buffer. Lowers to global_prefetch_b8.
    const size_t pidx = (row + 8 < rows) ? idx + (size_t)8 * (size_t)V4 : idx;
    __builtin_prefetch(&logits4[pidx], 0, 1);

    // Wave-uniform remaining-syllable count for this (s,b) row.
    const uint32_t rem  = (uint32_t)remain[row];           // 0..7
    const uint32_t bias = (127u - rem) * 0x01010101u;      // uniform SWAR bias

    // 4 packed syllable counts (each 1..7). byte > rem  <=>  bit7 of byte+bias.
    const uint32_t m = (pw2s4[v4] + bias) & 0x80808080u;

    v4f x = __builtin_nontemporal_load(&logits4[idx]);
    const float NEG_INF = -__builtin_inff();
    if (m & 0x00000080u) x.x = NEG_INF;
    if (m & 0x00008000u) x.y = NEG_INF;
    if (m & 0x00800000u) x.z = NEG_INF;
    if (m & 0x80000000u) x.w = NEG_INF;

    __builtin_nontemporal_store(x, &out4[idx]);
}

// ---------------------------------------------------------------------------
// Launch: inputs are (logits f32 [S,B,V], decoder_input i32 [S,B],
// word2syllables i32 [V]); output f32 [S,B,V].
// ---------------------------------------------------------------------------
extern "C" void kernel_launch(void* const* d_in, const int* in_sizes, int n_in,
                              void* d_out, int out_size, void* d_ws, size_t ws_size,
                              hipStream_t stream) {
    const float* logits = (const float*)d_in[0];
    const int*   dec    = (const int*)d_in[1];
    const int*   w2s    = (const int*)d_in[2];
    float*       out    = (float*)d_out;

    const int V    = in_sizes[2];          // 32000 (divisible by 4)
    const int rows = in_sizes[1];          // S*B = 4096
    const int B    = 128;
    const int S    = rows / B;
    const int V4   = V / 4;                // 8000 float4 per row

    // Workspace layout: [0, rows) remain bytes; then packed w2s bytes.
    uint8_t* remain = (uint8_t*)d_ws;
    uint8_t* packed = (uint8_t*)d_ws + (size_t)((rows + 15) & ~15);

    pack_w2s_kernel<<<(V + 255) / 256, 256, 0, stream>>>(w2s, packed, V);
    remain_scan_kernel<<<(B + 127) / 128, 128, 0, stream>>>(dec, w2s, remain, S, B);

    dim3 grid((V4 + 255) / 256, rows);
    mask_kernel<<<grid, 256, 0, stream>>>((const v4f*)logits,
                                          (const uint32_t*)packed,
                                          remain,
                                          (v4f*)out,
                                          V4, rows);
}